// GraphConstructor_12833362280663
// MI455X (gfx1250) — compile-verified
//
#include <hip/hip_runtime.h>
#include <math.h>

// ---------------- problem constants ----------------
#define N_NODES 5000
#define N_EDGES 160000
#define NHEADS  6
#define DHEAD   64
#define FDIM    384    // NHEADS*DHEAD
#define INDIM   131

typedef float v2f __attribute__((ext_vector_type(2)));
typedef float v4f __attribute__((ext_vector_type(4)));
typedef float v8f __attribute__((ext_vector_type(8)));

// ---- ordered-int encoding so atomicMax(int) == float max (handles negatives) ----
__device__ __forceinline__ int enc_f(float x) {
  int i = __float_as_int(x);
  return (i >= 0) ? i : (i ^ 0x7FFFFFFF);
}
__device__ __forceinline__ float dec_f(int i) {
  return __int_as_float((i >= 0) ? i : (i ^ 0x7FFFFFFF));
}

// ---------------- utility ----------------
__global__ void fill_kernel(float* __restrict__ p, float v, int n) {
  int i = blockIdx.x * blockDim.x + threadIdx.x;
  if (i < n) p[i] = v;
}

// ---------------- WMMA fp32 GEMM: C[M x 384] = A[M x K] * B[K x 384] ----------------
// One wave32 computes one 16x16 tile via V_WMMA_F32_16X16X4_F32.
// A frag (16x4): lanes 0-15 hold M=lane, VGPR0=K0,VGPR1=K1; lanes 16-31 hold K2/K3.
// B frag (4x16): mirrored (row K striped across lanes within a VGPR).
// C frag: VGPR v -> row (v + 8*half), col = lane&15.
// K is a compile-time constant: no bounds checks in the steady-state loop,
// A fragment is a single 8-byte load when K is even.
template <int K>
__global__ void gemm_wmma_f32(const float* __restrict__ A,
                              const float* __restrict__ B,
                              float* __restrict__ C, int M) {
  const int lane  = threadIdx.x & 31;
  const int wave  = threadIdx.x >> 5;            // 0..7
  const int mtile = blockIdx.x;                  // 16-row tile
  const int ntile = blockIdx.y * 8 + wave;       // 24 col-tiles total (grid.y=3)
  const int half  = lane >> 4;                   // K-half select
  const int m     = lane & 15;
  const int arow  = min(mtile * 16 + m, M - 1);  // clamp (stores are guarded)
  const int col   = ntile * 16 + m;

  const float* __restrict__ Arow = A + (long)arow * K + 2 * half; // element ka at [k0]
  const float* __restrict__ Bcol = B + (long)(2 * half) * FDIM + col;

  v8f acc = {0.f, 0.f, 0.f, 0.f, 0.f, 0.f, 0.f, 0.f};

  constexpr int KFULL = K & ~3;
#pragma unroll 4
  for (int k0 = 0; k0 < KFULL; k0 += 4) {
    v2f a, b;
    if constexpr ((K % 2) == 0) {
      a = *(const v2f*)(Arow + k0);              // global_load_b64 (8B aligned)
    } else {
      a.x = Arow[k0];
      a.y = Arow[k0 + 1];
    }
    b.x = Bcol[(long)k0 * FDIM];
    b.y = Bcol[(long)(k0 + 1) * FDIM];
    // 8 args: (neg_a, A, neg_b, B, c_mod, C, reuse_a, reuse_b)
    acc = __builtin_amdgcn_wmma_f32_16x16x4_f32(false, a, false, b, (short)0, acc,
                                                false, false);
  }
  if constexpr (KFULL < K) {                     // compile-time tail (K=131: 3 elems)
    const int ka = KFULL + 2 * half;
    v2f a = {0.f, 0.f}, b = {0.f, 0.f};
    if (ka < K)     { a.x = Arow[KFULL];     b.x = Bcol[(long)KFULL * FDIM]; }
    if (ka + 1 < K) { a.y = Arow[KFULL + 1]; b.y = Bcol[(long)(KFULL + 1) * FDIM]; }
    acc = __builtin_amdgcn_wmma_f32_16x16x4_f32(false, a, false, b, (short)0, acc,
                                                false, false);
  }

#pragma unroll
  for (int v = 0; v < 8; ++v) {
    int mr = mtile * 16 + v + 8 * half;
    if (mr < M) C[(long)mr * FDIM + col] = acc[v];
  }
}

// ---------------- el/er per (node, head) + softmax-state init ----------------
__global__ void attn_coef_kernel(const float* __restrict__ f,
                                 const float* __restrict__ al,
                                 const float* __restrict__ ar,
                                 float* __restrict__ el, float* __restrict__ er,
                                 int* __restrict__ emax_enc, float* __restrict__ den) {
  int i = blockIdx.x * blockDim.x + threadIdx.x;   // n*NHEADS + h
  if (i >= N_NODES * NHEADS) return;
  int n = i / NHEADS, h = i % NHEADS;
  const float* fp  = f  + (long)n * FDIM + h * DHEAD;
  const float* alp = al + h * DHEAD;
  const float* arp = ar + h * DHEAD;
  float sl = 0.f, sr = 0.f;
#pragma unroll 8
  for (int d = 0; d < DHEAD; ++d) { float x = fp[d]; sl += x * alp[d]; sr += x * arp[d]; }
  el[i] = sl;
  er[i] = sr;
  emax_enc[i] = enc_f(-INFINITY);
  den[i] = 0.f;
}

// ---------------- edge pass 1: e = leakyrelu(el[src]+er[dst]); atomic max per dst ----------------
__global__ void edge_pass1(const int* __restrict__ src, const int* __restrict__ dst,
                           const float* __restrict__ el, const float* __restrict__ er,
                           float* __restrict__ e_edge, int* __restrict__ emax_enc) {
  int e = blockIdx.x * blockDim.x + threadIdx.x;
  if (e >= N_EDGES) return;
  int s = src[e], d = dst[e];
#pragma unroll
  for (int h = 0; h < NHEADS; ++h) {
    float v = el[s * NHEADS + h] + er[d * NHEADS + h];
    v = (v > 0.f) ? v : 0.2f * v;                 // leaky_relu(0.2)
    e_edge[(long)e * NHEADS + h] = v;
    atomicMax(&emax_enc[d * NHEADS + h], enc_f(v));
  }
}

// ---------------- edge pass 2: ex = exp(e - emax[dst]); atomic add to den ----------------
__global__ void edge_pass2(const int* __restrict__ dst,
                           float* __restrict__ e_edge,
                           const int* __restrict__ emax_enc,
                           float* __restrict__ den) {
  int e = blockIdx.x * blockDim.x + threadIdx.x;
  if (e >= N_EDGES) return;
  int d = dst[e];
#pragma unroll
  for (int h = 0; h < NHEADS; ++h) {
    float m  = dec_f(emax_enc[d * NHEADS + h]);
    float ex = __expf(e_edge[(long)e * NHEADS + h] - m);
    e_edge[(long)e * NHEADS + h] = ex;            // store exp in place
    atomicAdd(&den[d * NHEADS + h], ex);
  }
}

// ---------------- edge pass 3: out[dst] += f[src] * alpha ----------------
__global__ void edge_pass3(const int* __restrict__ src, const int* __restrict__ dst,
                           const float* __restrict__ e_edge,
                           const float* __restrict__ den,
                           const float* __restrict__ f,
                           float* __restrict__ out) {
  int t = blockIdx.x * blockDim.x + threadIdx.x;  // edge*NHEADS + h
  if (t >= N_EDGES * NHEADS) return;
  int e = t / NHEADS, h = t % NHEADS;
  int s = src[e], d = dst[e];
  float alpha = e_edge[(long)e * NHEADS + h] / den[d * NHEADS + h];
  const float* fp = f   + (long)s * FDIM + h * DHEAD;
  float*       op = out + (long)d * FDIM + h * DHEAD;
#pragma unroll 8
  for (int k = 0; k < DHEAD; ++k) atomicAdd(&op[k], fp[k] * alpha);
}

// ---------------- bias + optional ELU (in place) ----------------
__global__ void finish_kernel(float* __restrict__ out, const float* __restrict__ b,
                              int activate) {
  int i = blockIdx.x * blockDim.x + threadIdx.x;
  if (i >= N_NODES * FDIM) return;
  float v = out[i] + b[i % FDIM];
  if (activate) v = (v > 0.f) ? v : expm1f(v);    // elu(alpha=1)
  out[i] = v;
}

// ---------------- emb = mean over heads ----------------
__global__ void emb_kernel(const float* __restrict__ f3, float* __restrict__ emb) {
  int i = blockIdx.x * blockDim.x + threadIdx.x;  // n*DHEAD + d
  if (i >= N_NODES * DHEAD) return;
  int n = i / DHEAD, d = i % DHEAD;
  const float* fp = f3 + (long)n * FDIM + d;
  float s = 0.f;
#pragma unroll
  for (int h = 0; h < NHEADS; ++h) s += fp[h * DHEAD];
  emb[i] = s * (1.0f / NHEADS);
}

// ---------------- s1 = emb @ pw[:64], s2 = emb @ pw[64:128] ----------------
__global__ void score_kernel(const float* __restrict__ emb, const float* __restrict__ pw,
                             float* __restrict__ s1, float* __restrict__ s2) {
  int n = blockIdx.x * blockDim.x + threadIdx.x;
  if (n >= N_NODES) return;
  const float* ep = emb + (long)n * DHEAD;
  float a = 0.f, b = 0.f;
#pragma unroll 8
  for (int d = 0; d < DHEAD; ++d) { float x = ep[d]; a += x * pw[d]; b += x * pw[DHEAD + d]; }
  s1[n] = a;
  s2[n] = b;
}

// ---------------- OD[i,j] = tanh(s2[i] + s1[j] + dis[i,j]*pw[128] + pb) ----------------
// Streaming kernel: 100MB read + 100MB write with zero reuse -> non-temporal b128
// traffic so the 192MB L2 isn't churned. N_NODES % 4 == 0, so a flat float4 index
// never straddles a row.
__global__ void od_kernel4(const float* __restrict__ dis, const float* __restrict__ s1,
                           const float* __restrict__ s2, const float* __restrict__ pw,
                           const float* __restrict__ pb, float* __restrict__ od) {
  long t = (long)blockIdx.x * blockDim.x + threadIdx.x;  // float4 index
  const long total4 = (long)N_NODES * N_NODES / 4;
  if (t >= total4) return;
  long base = 4 * t;
  int i = (int)(base / N_NODES);
  int j = (int)(base % N_NODES);
  float w = pw[2 * DHEAD];
  float c = s2[i] + pb[0];
  v4f d = __builtin_nontemporal_load((const v4f*)(dis + base));
  v4f r;
  r.x = tanhf(c + s1[j + 0] + d.x * w);
  r.y = tanhf(c + s1[j + 1] + d.y * w);
  r.z = tanhf(c + s1[j + 2] + d.z * w);
  r.w = tanhf(c + s1[j + 3] + d.w * w);
  __builtin_nontemporal_store(r, (v4f*)(od + base));
}

// =====================================================================
extern "C" void kernel_launch(void* const* d_in, const int* in_sizes, int n_in,
                              void* d_out, int out_size, void* d_ws, size_t ws_size,
                              hipStream_t stream) {
  const float* nfeats = (const float*)d_in[0];
  const float* dis    = (const float*)d_in[1];
  const int*   src    = (const int*)d_in[2];
  const int*   dst    = (const int*)d_in[3];
  const float* pw     = (const float*)d_in[20];
  const float* pb     = (const float*)d_in[21];
  float* od = (float*)d_out;

  // workspace layout (floats)
  float* ws = (float*)d_ws;
  size_t o = 0;
  float* bufA   = ws + o; o += (size_t)N_NODES * FDIM;   // f (post-GEMM features)
  float* bufB   = ws + o; o += (size_t)N_NODES * FDIM;   // h / aggregation output
  float* el     = ws + o; o += (size_t)N_NODES * NHEADS;
  float* er     = ws + o; o += (size_t)N_NODES * NHEADS;
  int*   emaxE  = (int*)(ws + o); o += (size_t)N_NODES * NHEADS;
  float* den    = ws + o; o += (size_t)N_NODES * NHEADS;
  float* e_edge = ws + o; o += (size_t)N_EDGES * NHEADS;
  float* emb    = ws + o; o += (size_t)N_NODES * DHEAD;
  float* s1     = ws + o; o += (size_t)N_NODES;
  float* s2     = ws + o; o += (size_t)N_NODES;
  (void)ws_size; (void)n_in; (void)in_sizes; (void)out_size;

  const int BLK = 256;
  const dim3 gemm_grid((N_NODES + 15) / 16, (FDIM / 16) / 8);  // 313 x 3, 8 waves/block
  const int nh_blocks   = (N_NODES * NHEADS + BLK - 1) / BLK;
  const int edge_blocks = (N_EDGES + BLK - 1) / BLK;
  const int eh_blocks   = (N_EDGES * NHEADS + BLK - 1) / BLK;
  const int nf_blocks   = (N_NODES * FDIM + BLK - 1) / BLK;

  for (int l = 0; l < 4; ++l) {
    const float* h_in = (l == 0) ? nfeats : bufB;
    const float* Wl   = (const float*)d_in[4 + 4 * l];
    const float* all_ = (const float*)d_in[5 + 4 * l];
    const float* arl  = (const float*)d_in[6 + 4 * l];
    const float* bl   = (const float*)d_in[7 + 4 * l];
    const int activate = (l < 3) ? 1 : 0;

    // f = h @ W  (WMMA fp32 path; K specialized at compile time)
    if (l == 0)
      gemm_wmma_f32<INDIM><<<gemm_grid, BLK, 0, stream>>>(h_in, Wl, bufA, N_NODES);
    else
      gemm_wmma_f32<FDIM><<<gemm_grid, BLK, 0, stream>>>(h_in, Wl, bufA, N_NODES);
    // attention coefficients + per-(dst,head) softmax state init
    attn_coef_kernel<<<nh_blocks, BLK, 0, stream>>>(bufA, all_, arl, el, er, emaxE, den);
    // zero aggregation target (bufB already consumed by the GEMM)
    fill_kernel<<<nf_blocks, BLK, 0, stream>>>(bufB, 0.f, N_NODES * FDIM);
    // edge softmax (max, exp+sum, weighted scatter)
    edge_pass1<<<edge_blocks, BLK, 0, stream>>>(src, dst, el, er, e_edge, emaxE);
    edge_pass2<<<edge_blocks, BLK, 0, stream>>>(dst, e_edge, emaxE, den);
    edge_pass3<<<eh_blocks, BLK, 0, stream>>>(src, dst, e_edge, den, bufA, bufB);
    // bias + ELU
    finish_kernel<<<nf_blocks, BLK, 0, stream>>>(bufB, bl, activate);
  }

  // head-mean embedding, pair scores, pairwise tanh output
  emb_kernel<<<(N_NODES * DHEAD + BLK - 1) / BLK, BLK, 0, stream>>>(bufB, emb);
  score_kernel<<<(N_NODES + BLK - 1) / BLK, BLK, 0, stream>>>(emb, pw, s1, s2);
  const long total4 = (long)N_NODES * N_NODES / 4;
  od_kernel4<<<(int)((total4 + BLK - 1) / BLK), BLK, 0, stream>>>(dis, s1, s2, pw, pb, od);
}